// GCNRecommender_37546604102312
// MI455X (gfx1250) — compile-verified
//
#include <hip/hip_runtime.h>

// ---------------------------------------------------------------------------
// GCNRecommender on MI455X (gfx1250): fp32 WMMA (V_WMMA_F32_16X16X4_F32) for
// all dense GEMMs, float4 gather + global_atomic_add_f32 for scatter-mean.
// Weights are pre-packed into the WMMA B-operand register layout so the GEMM
// inner loop issues a single coalesced b64 load per 16x16x4 step.
// ---------------------------------------------------------------------------

typedef __attribute__((ext_vector_type(2))) float v2f;
typedef __attribute__((ext_vector_type(8))) float v8f;

enum : int { NU = 200000, NI = 100000, E_UU = 1000000, E_UI = 2000000,
             DU = 128, DI = 64, H = 64, O = 32 };

// ---------------------------------------------------------------------------
// Zero-fill (float4 grid-stride). Workspace is poisoned, so sums/counts must
// be zeroed every call.
// ---------------------------------------------------------------------------
__global__ __launch_bounds__(256) void zero_kernel(float4* __restrict__ p, size_t n4) {
  size_t i = (size_t)blockIdx.x * blockDim.x + threadIdx.x;
  size_t stride = (size_t)gridDim.x * blockDim.x;
  float4 z = make_float4(0.f, 0.f, 0.f, 0.f);
  for (; i < n4; i += stride) p[i] = z;
}

// ---------------------------------------------------------------------------
// Pack W[K,N] (row-major, optionally summed with Wb) into the WMMA fp32
// B-operand layout:  P2[(s*N + col)*2 + half] = { W[4s+2h, col], W[4s+2h+1, col] }
// so lane (col = lane&15 + 16j, half = lane>>4) loads one contiguous float2,
// and the whole wave covers a contiguous 256B line per n-tile per k-step.
// ---------------------------------------------------------------------------
__global__ __launch_bounds__(256) void pack_w_kernel(const float* __restrict__ Wa,
                                                     const float* __restrict__ Wb,
                                                     float2* __restrict__ P,
                                                     int K, int N) {
  int i = blockIdx.x * blockDim.x + threadIdx.x;   // over (K/4)*N*2 float2 slots
  int total = (K >> 2) * N * 2;
  if (i >= total) return;
  int half = i & 1;
  int cs = i >> 1;
  int col = cs % N;
  int s = cs / N;
  int k = 4 * s + 2 * half;
  size_t i0 = (size_t)k * N + col;
  size_t i1 = (size_t)(k + 1) * N + col;
  float x = Wa[i0], y = Wa[i1];
  if (Wb) { x += Wb[i0]; y += Wb[i1]; }
  P[i] = make_float2(x, y);
}

// ---------------------------------------------------------------------------
// Segment-sum scatter: 16 lanes per edge, each lane moves a float4 of the
// 64-float feature row; lane 0 bumps the degree count (layer 1 only; degrees
// are identical in layer 2 so counts are reused).
// ---------------------------------------------------------------------------
__global__ __launch_bounds__(256) void scatter_kernel(const float* __restrict__ x,
                                                      const int* __restrict__ src,
                                                      const int* __restrict__ dst, int E,
                                                      float* __restrict__ sum,
                                                      float* __restrict__ cnt) {
  size_t t = (size_t)blockIdx.x * blockDim.x + threadIdx.x;
  int e = (int)(t >> 4);
  int l = (int)(t & 15);
  if (e >= E) return;
  int s = src[e];
  int d = dst[e];
  const float4 v = ((const float4*)(x + (size_t)s * H))[l];
  float* o = sum + (size_t)d * H + l * 4;
  atomicAdd(o + 0, v.x);
  atomicAdd(o + 1, v.y);
  atomicAdd(o + 2, v.z);
  atomicAdd(o + 3, v.w);
  if (cnt != nullptr && l == 0) atomicAdd(cnt + d, 1.0f);
}

// ---------------------------------------------------------------------------
// Fused multi-input WMMA GEMM:
//   Out = act( scale * ( sum_p mean_p(A_p) @ W_p  +  bias0 (+bias1) ) )
// mean_p(A)[m,k] = A[m,k] / max(cnt_p[m],1) when cnt_p != null, else A[m,k].
// A_p: [M,K] row-major; Wp_p: pre-packed B layout (pack_w_kernel); N=NTILES*16.
// One wave owns a 16 x N output tile; 8 waves/block -> 128 rows/block.
// fp32 WMMA layouts (ISA 7.12.2):
//   A 16x4 : lane = 16*(k>>1) + m, vgpr j -> k = 2*(lane>>4)+j
//   B 4x16 : lane = 16*(k>>1) + n, vgpr j -> k = 2*(lane>>4)+j
//   C 16x16: vgpr q, col = lane&15, row = 8*(lane>>4)+q
// ---------------------------------------------------------------------------
struct GemmPair { const float* A; const float* cnt; const float* Wp; };

template <int NTILES>
__device__ __forceinline__ void gemm_accum(v8f acc[NTILES], const float* __restrict__ A,
                                           const float* __restrict__ cnt,
                                           const float* __restrict__ Wp,
                                           int mload, int K, int lane) {
  const int N = NTILES * 16;
  const int r = lane & 15;
  const int kk = (lane >> 4) * 2;
  const int half = lane >> 4;
  float rs = 1.0f;
  if (cnt) rs = 1.0f / fmaxf(cnt[mload], 1.0f);
  const float* arow = A + (size_t)mload * K;
  const v2f* P = (const v2f*)Wp;
  for (int k = 0; k < K; k += 4) {
    v2f a = *(const v2f*)(arow + k + kk);   // 8B-aligned packed load
    a.x *= rs;
    a.y *= rs;
    const int s = k >> 2;
#pragma unroll
    for (int j = 0; j < NTILES; ++j) {
      v2f b = P[((size_t)s * N + j * 16 + r) * 2 + half];  // one b64, wave-coalesced
      acc[j] = __builtin_amdgcn_wmma_f32_16x16x4_f32(
          false, a, false, b, (short)0, acc[j], false, false);
    }
  }
}

template <int NTILES>
__global__ __launch_bounds__(256) void sage_gemm_kernel(
    GemmPair p0, GemmPair p1, GemmPair p2, GemmPair p3, int npairs,
    int M, int K, const float* __restrict__ bias0, const float* __restrict__ bias1,
    float scale, int relu, float* __restrict__ Out) {
  const int N = NTILES * 16;
  const int lane = threadIdx.x & 31;
  const int wave = threadIdx.x >> 5;
  const int m0 = blockIdx.x * 128 + wave * 16;   // wave-uniform
  if (m0 >= M) return;                            // uniform branch: EXEC stays full
  const int r = lane & 15;
  int mload = m0 + r;
  if (mload > M - 1) mload = M - 1;               // clamp tail reads

  v8f acc[NTILES];
#pragma unroll
  for (int j = 0; j < NTILES; ++j)
#pragma unroll
    for (int q = 0; q < 8; ++q) acc[j][q] = 0.0f;

  if (npairs > 0) gemm_accum<NTILES>(acc, p0.A, p0.cnt, p0.Wp, mload, K, lane);
  if (npairs > 1) gemm_accum<NTILES>(acc, p1.A, p1.cnt, p1.Wp, mload, K, lane);
  if (npairs > 2) gemm_accum<NTILES>(acc, p2.A, p2.cnt, p2.Wp, mload, K, lane);
  if (npairs > 3) gemm_accum<NTILES>(acc, p3.A, p3.cnt, p3.Wp, mload, K, lane);

  const int mBase = (lane >> 4) * 8;
#pragma unroll
  for (int j = 0; j < NTILES; ++j) {
    const int col = j * 16 + r;                   // all 8 acc elems share this column
    float b = 0.0f;
    if (bias0) b = bias0[col];
    if (bias1) b += bias1[col];
#pragma unroll
    for (int q = 0; q < 8; ++q) {
      int row = m0 + mBase + q;
      if (row < M) {
        float v = (acc[j][q] + b) * scale;
        if (relu) v = fmaxf(v, 0.0f);
        Out[(size_t)row * N + col] = v;
      }
    }
  }
}

// ---------------------------------------------------------------------------
extern "C" void kernel_launch(void* const* d_in, const int* in_sizes, int n_in,
                              void* d_out, int out_size, void* d_ws, size_t ws_size,
                              hipStream_t stream) {
  const float* x_user = (const float*)d_in[0];
  const float* x_item = (const float*)d_in[1];
  const int* soc_src = (const int*)d_in[2];
  const int* soc_dst = (const int*)d_in[3];
  const int* int_src = (const int*)d_in[4];
  const int* int_dst = (const int*)d_in[5];
  const float* Wpu = (const float*)d_in[6];
  const float* bpu = (const float*)d_in[7];
  const float* Wpi = (const float*)d_in[8];
  const float* bpi = (const float*)d_in[9];
  const float* W1l_uu = (const float*)d_in[10]; const float* b1l_uu = (const float*)d_in[11];
  const float* W1r_uu = (const float*)d_in[12];
  const float* W1l_ui = (const float*)d_in[13]; const float* b1l_ui = (const float*)d_in[14];
  const float* W1r_ui = (const float*)d_in[15];
  const float* W1l_iu = (const float*)d_in[16]; const float* b1l_iu = (const float*)d_in[17];
  const float* W1r_iu = (const float*)d_in[18];
  const float* W2l_uu = (const float*)d_in[19]; const float* b2l_uu = (const float*)d_in[20];
  const float* W2r_uu = (const float*)d_in[21];
  const float* W2l_ui = (const float*)d_in[22]; const float* b2l_ui = (const float*)d_in[23];
  const float* W2r_ui = (const float*)d_in[24];
  const float* W2l_iu = (const float*)d_in[25]; const float* b2l_iu = (const float*)d_in[26];
  const float* W2r_iu = (const float*)d_in[27];

  // ---- workspace layout (floats). Sums first, then counts (zeroed together),
  //      then persistent activations + packed-weight scratch.
  float* ws = (float*)d_ws;
  size_t off = 0;
  float* sum_uu = ws + off; off += (size_t)NU * H;
  float* sum_iu = ws + off; off += (size_t)NU * H;
  float* sum_ui = ws + off; off += (size_t)NI * H;
  const size_t sums_elems = off;                 // re-zeroed before layer 2
  float* cnt_uu = ws + off; off += NU;
  float* cnt_iu = ws + off; off += NU;
  float* cnt_ui = ws + off; off += NI;
  const size_t zero_elems = off;                 // zeroed once per call
  float* hu  = ws + off; off += (size_t)NU * H;
  float* hi  = ws + off; off += (size_t)NI * H;
  float* h1u = ws + off; off += (size_t)NU * H;
  float* h1i = ws + off; off += (size_t)NI * H;
  // packed weights (same element count as the source matrices)
  float* pWpu    = ws + off; off += (size_t)DU * H;
  float* pWpi    = ws + off; off += (size_t)DI * H;
  float* pW1l_uu = ws + off; off += (size_t)H * H;
  float* pW1l_iu = ws + off; off += (size_t)H * H;
  float* pWr1sum = ws + off; off += (size_t)H * H;   // W1r_uu + W1r_iu, packed
  float* pW1l_ui = ws + off; off += (size_t)H * H;
  float* pW1r_ui = ws + off; off += (size_t)H * H;
  float* pW2l_uu = ws + off; off += (size_t)H * O;
  float* pW2l_iu = ws + off; off += (size_t)H * O;
  float* pWr2sum = ws + off; off += (size_t)H * O;   // W2r_uu + W2r_iu, packed
  float* pW2l_ui = ws + off; off += (size_t)H * O;
  float* pW2r_ui = ws + off; off += (size_t)H * O;

  float* h2u = (float*)d_out;                    // [NU, O]
  float* h2i = (float*)d_out + (size_t)NU * O;   // [NI, O]

  const GemmPair PNUL{nullptr, nullptr, nullptr};
  const dim3 blk(256);
  const int gridU = (NU + 127) / 128;
  const int gridI = (NI + 127) / 128;
  const unsigned gs_uu = (unsigned)(((size_t)E_UU * 16 + 255) / 256);
  const unsigned gs_ui = (unsigned)(((size_t)E_UI * 16 + 255) / 256);

  // 0) zero sums + counts; pack all weights into WMMA B layout
  zero_kernel<<<2048, blk, 0, stream>>>((float4*)ws, zero_elems / 4);
  {
    const int gHH = (H * H / 2 + 255) / 256;     // (K/4)*N*2 = K*N/2 slots
    const int gHO = (H * O / 2 + 255) / 256;
    pack_w_kernel<<<(DU * H / 2 + 255) / 256, blk, 0, stream>>>(Wpu, nullptr, (float2*)pWpu, DU, H);
    pack_w_kernel<<<(DI * H / 2 + 255) / 256, blk, 0, stream>>>(Wpi, nullptr, (float2*)pWpi, DI, H);
    pack_w_kernel<<<gHH, blk, 0, stream>>>(W1l_uu, nullptr, (float2*)pW1l_uu, H, H);
    pack_w_kernel<<<gHH, blk, 0, stream>>>(W1l_iu, nullptr, (float2*)pW1l_iu, H, H);
    pack_w_kernel<<<gHH, blk, 0, stream>>>(W1r_uu, W1r_iu, (float2*)pWr1sum, H, H);
    pack_w_kernel<<<gHH, blk, 0, stream>>>(W1l_ui, nullptr, (float2*)pW1l_ui, H, H);
    pack_w_kernel<<<gHH, blk, 0, stream>>>(W1r_ui, nullptr, (float2*)pW1r_ui, H, H);
    pack_w_kernel<<<gHO, blk, 0, stream>>>(W2l_uu, nullptr, (float2*)pW2l_uu, H, O);
    pack_w_kernel<<<gHO, blk, 0, stream>>>(W2l_iu, nullptr, (float2*)pW2l_iu, H, O);
    pack_w_kernel<<<gHO, blk, 0, stream>>>(W2r_uu, W2r_iu, (float2*)pWr2sum, H, O);
    pack_w_kernel<<<gHO, blk, 0, stream>>>(W2l_ui, nullptr, (float2*)pW2l_ui, H, O);
    pack_w_kernel<<<gHO, blk, 0, stream>>>(W2r_ui, nullptr, (float2*)pW2r_ui, H, O);
  }

  // 1) input projections: hu = x_user@Wpu + bpu ; hi = x_item@Wpi + bpi
  sage_gemm_kernel<4><<<gridU, blk, 0, stream>>>(
      GemmPair{x_user, nullptr, pWpu}, PNUL, PNUL, PNUL, 1, NU, DU, bpu, nullptr, 1.0f, 0, hu);
  sage_gemm_kernel<4><<<gridI, blk, 0, stream>>>(
      GemmPair{x_item, nullptr, pWpi}, PNUL, PNUL, PNUL, 1, NI, DI, bpi, nullptr, 1.0f, 0, hi);

  // 2) layer-1 segment sums (+degree counts, reused for layer 2)
  scatter_kernel<<<gs_uu, blk, 0, stream>>>(hu, soc_src, soc_dst, E_UU, sum_uu, cnt_uu);
  scatter_kernel<<<gs_ui, blk, 0, stream>>>(hu, int_src, int_dst, E_UI, sum_ui, cnt_ui);
  scatter_kernel<<<gs_ui, blk, 0, stream>>>(hi, int_dst, int_src, E_UI, sum_iu, cnt_iu);

  // 3) layer-1 combine
  //    h1u = relu(0.5*(mean_uu@W1l_uu + mean_iu@W1l_iu + hu@(W1r_uu+W1r_iu) + b1l_uu + b1l_iu))
  sage_gemm_kernel<4><<<gridU, blk, 0, stream>>>(
      GemmPair{sum_uu, cnt_uu, pW1l_uu}, GemmPair{sum_iu, cnt_iu, pW1l_iu},
      GemmPair{hu, nullptr, pWr1sum}, PNUL, 3, NU, H, b1l_uu, b1l_iu, 0.5f, 1, h1u);
  //    h1i = relu(mean_ui@W1l_ui + hi@W1r_ui + b1l_ui)
  sage_gemm_kernel<4><<<gridI, blk, 0, stream>>>(
      GemmPair{sum_ui, cnt_ui, pW1l_ui}, GemmPair{hi, nullptr, pW1r_ui},
      PNUL, PNUL, 2, NI, H, b1l_ui, nullptr, 1.0f, 1, h1i);

  // 4) re-zero sums (keep counts), layer-2 segment sums
  zero_kernel<<<2048, blk, 0, stream>>>((float4*)ws, sums_elems / 4);
  scatter_kernel<<<gs_uu, blk, 0, stream>>>(h1u, soc_src, soc_dst, E_UU, sum_uu, nullptr);
  scatter_kernel<<<gs_ui, blk, 0, stream>>>(h1u, int_src, int_dst, E_UI, sum_ui, nullptr);
  scatter_kernel<<<gs_ui, blk, 0, stream>>>(h1i, int_dst, int_src, E_UI, sum_iu, nullptr);

  // 5) layer-2 combine (N = 32), written straight to d_out
  sage_gemm_kernel<2><<<gridU, blk, 0, stream>>>(
      GemmPair{sum_uu, cnt_uu, pW2l_uu}, GemmPair{sum_iu, cnt_iu, pW2l_iu},
      GemmPair{h1u, nullptr, pWr2sum}, PNUL, 3, NU, H, b2l_uu, b2l_iu, 0.5f, 1, h2u);
  sage_gemm_kernel<2><<<gridI, blk, 0, stream>>>(
      GemmPair{sum_ui, cnt_ui, pW2l_ui}, GemmPair{h1i, nullptr, pW2r_ui},
      PNUL, PNUL, 2, NI, H, b2l_ui, nullptr, 1.0f, 1, h2i);
}